// WeightedPeakAwareLoss_71545565217535
// MI455X (gfx1250) — compile-verified
//
#include <hip/hip_runtime.h>
#include <hip/hip_bf16.h>

#define HOURS 24
#define DAYS  168
#define SEQ   (HOURS * DAYS)   // 4032
#define CHUNKS (SEQ / 4)       // 1008 float4 per row
#define BLK1  256
#define BLK2  1024
#define SORTN 4096

typedef __attribute__((ext_vector_type(2))) float v2f;
typedef __attribute__((ext_vector_type(8))) float v8f;
typedef __attribute__((ext_vector_type(4))) int   v4i;

#define AS1 __attribute__((address_space(1)))
#define AS3 __attribute__((address_space(3)))

#if __has_builtin(__builtin_amdgcn_global_load_async_to_lds_b128)
#define USE_ASYNC 1
#else
#define USE_ASYNC 0
#endif

__device__ __forceinline__ float reduce16(float t) {
  // sum across a 16-lane group (lanes 0..15 of D vgpr0 hold the 16 column sums)
  t += __shfl_xor(t, 1, 16);
  t += __shfl_xor(t, 2, 16);
  t += __shfl_xor(t, 4, 16);
  t += __shfl_xor(t, 8, 16);
  return t;
}

// One workgroup per sample: single global pass -> 5 per-sample scalars.
__global__ __launch_bounds__(BLK1) void wpal_per_sample(
    const float* __restrict__ y_pred, const float* __restrict__ y_true,
    float* __restrict__ o_wsq, float* __restrict__ o_ws,
    float* __restrict__ o_psq, float* __restrict__ o_avg,
    float* __restrict__ o_dt)
{
  __shared__ float s_true[SEQ];
  __shared__ float s_pred[SEQ];
  __shared__ int   s_tt[DAYS];
  __shared__ float s_red[5 * BLK1];

  const int tid = threadIdx.x;
  const int b = blockIdx.x;
  const float* gt = y_true + (size_t)b * SEQ;
  const float* gp = y_pred + (size_t)b * SEQ;

  // ---- Stage both rows into LDS (async global->LDS, tracked by ASYNCcnt) ----
#if USE_ASYNC
  for (int c = tid; c < CHUNKS; c += BLK1) {
    __builtin_amdgcn_global_load_async_to_lds_b128(
        (AS1 v4i*)(gt + c * 4), (AS3 v4i*)(s_true + c * 4), 0, 0);
    __builtin_amdgcn_global_load_async_to_lds_b128(
        (AS1 v4i*)(gp + c * 4), (AS3 v4i*)(s_pred + c * 4), 0, 0);
  }
#if __has_builtin(__builtin_amdgcn_s_wait_asynccnt)
  __builtin_amdgcn_s_wait_asynccnt(0);
#else
  asm volatile("s_wait_asynccnt 0" ::: "memory");
#endif
#else
  for (int c = tid; c < CHUNKS; c += BLK1) {
    ((float4*)s_true)[c] = ((const float4*)gt)[c];
    ((float4*)s_pred)[c] = ((const float4*)gp)[c];
  }
#endif
  __syncthreads();

  // ---- Phase A: per-day daytime-masked argmax/peaks (thread == day) ----
  float a_psq = 0.f, a_pk = 0.f, a_dt = 0.f;
  if (tid < DAYS) {
    const int base = tid * HOURS;
    float tpk = -1e30f, ppk = -1e30f;
    int tt = 6, pt = 6;
#pragma unroll
    for (int h = 6; h <= 20; ++h) {   // argmax keeps FIRST max -> strict '>'
      float tv = s_true[base + h];
      float pv = s_pred[base + h];
      if (tv > tpk) { tpk = tv; tt = h; }
      if (pv > ppk) { ppk = pv; pt = h; }
    }
    s_tt[tid] = tt;
    float dpk = ppk - tpk;
    a_psq = dpk * dpk;
    a_pk  = tpk;
    a_dt  = fabsf((float)(pt - tt));
  }
  __syncthreads();

  // ---- Phase B: weighted squared error over all 4032 elements ----
  float a_wsq = 0.f, a_ws = 0.f;
  for (int i = tid; i < SEQ; i += BLK1) {
    int day = i / HOURS;
    int h   = i - day * HOURS;
    float w = ((h >= 6) && (h <= 20)) ? 2.0f : 0.5f;
    int tt = s_tt[day];
    if ((h >= tt - 2) && (h <= tt + 2)) w *= 3.0f;
    float d = s_pred[i] - s_true[i];
    a_wsq += w * d * d;
    a_ws  += w;
  }

  s_red[0 * BLK1 + tid] = a_wsq;
  s_red[1 * BLK1 + tid] = a_ws;
  s_red[2 * BLK1 + tid] = a_psq;
  s_red[3 * BLK1 + tid] = a_pk;
  s_red[4 * BLK1 + tid] = a_dt;
  __syncthreads();

  // ---- WMMA block reduction: wave 0 sums 256 values per accumulator.
  // D = A(ones,16x4) x B(4x16) + C  =>  D[m,n] = sum_k B[k,n]; summing the 16
  // column sums yields the exact total independent of B's lane mapping.
  if (tid < 32) {
    const int lane = tid;
    v2f ones; ones.x = 1.0f; ones.y = 1.0f;
    float tot0 = 0.f, tot1 = 0.f, tot2 = 0.f, tot3 = 0.f, tot4 = 0.f;
#pragma unroll
    for (int a = 0; a < 5; ++a) {
      v8f acc = {0.f, 0.f, 0.f, 0.f, 0.f, 0.f, 0.f, 0.f};
#pragma unroll
      for (int c = 0; c < 4; ++c) {   // 4 chunks of 64 values, C-accumulated
        v2f bv;
        bv.x = s_red[a * BLK1 + c * 64 + lane];
        bv.y = s_red[a * BLK1 + c * 64 + 32 + lane];
        acc = __builtin_amdgcn_wmma_f32_16x16x4_f32(
            false, ones, false, bv, (short)0, acc, false, false);
      }
      float t = reduce16(acc[0]);
      if (a == 0) tot0 = t; else if (a == 1) tot1 = t; else if (a == 2) tot2 = t;
      else if (a == 3) tot3 = t; else tot4 = t;
    }
    if (tid == 0) {
      o_wsq[b] = tot0;
      o_ws[b]  = tot1;
      o_psq[b] = tot2;
      o_avg[b] = tot3 * (1.0f / (float)DAYS);
      o_dt[b]  = tot4;
    }
  }
}

// Single block: quantile via bitonic sort + final deterministic reductions.
__global__ __launch_bounds__(BLK2) void wpal_finalize(
    const float* __restrict__ i_wsq, const float* __restrict__ i_ws,
    const float* __restrict__ i_psq, const float* __restrict__ i_avg,
    const float* __restrict__ i_dt, float* __restrict__ out, int B)
{
  __shared__ float s_sort[SORTN];
  __shared__ float s_red[BLK2];
  __shared__ float s_tot[5];
  __shared__ float s_thr;

  const int tid = threadIdx.x;
  for (int i = tid; i < SORTN; i += BLK2)
    s_sort[i] = (i < B) ? i_avg[i] : 3.0e38f;   // pad high: keeps low order stats
  __syncthreads();

  // Bitonic sort (ascending), 4096 elements, 1024 threads
  for (int k = 2; k <= SORTN; k <<= 1) {
    for (int j = k >> 1; j > 0; j >>= 1) {
      for (int i = tid; i < SORTN; i += BLK2) {
        int ixj = i ^ j;
        if (ixj > i) {
          float a = s_sort[i], c = s_sort[ixj];
          bool up = ((i & k) == 0);
          if ((a > c) == up) { s_sort[i] = c; s_sort[ixj] = a; }
        }
      }
      __syncthreads();
    }
  }

  if (tid == 0) {
    // jnp.quantile 'linear': idx = q*(B-1)
    float q = 0.75f * (float)(B - 1);
    int i0 = (int)q;
    float frac = q - (float)i0;
    float lo = s_sort[i0];
    float hi = (i0 + 1 < B) ? s_sort[i0 + 1] : lo;
    s_thr = lo + frac * (hi - lo);
  }
  __syncthreads();
  const float thr = s_thr;

  float acc0 = 0.f, acc1 = 0.f, acc2 = 0.f, acc3 = 0.f, acc4 = 0.f;
  for (int s = tid; s < B; s += BLK2) {
    float w = (i_avg[s] > thr) ? 1.5f : 1.0f;
    acc0 += i_wsq[s];
    acc1 += i_ws[s];
    acc2 += w * i_psq[s];
    acc3 += w;
    acc4 += i_dt[s];
  }

#pragma unroll
  for (int a = 0; a < 5; ++a) {
    float v = (a == 0) ? acc0 : (a == 1) ? acc1 : (a == 2) ? acc2
              : (a == 3) ? acc3 : acc4;
    __syncthreads();
    s_red[tid] = v;
    __syncthreads();
    for (int off = BLK2 / 2; off > 0; off >>= 1) {
      if (tid < off) s_red[tid] += s_red[tid + off];
      __syncthreads();
    }
    if (tid == 0) s_tot[a] = s_red[0];
  }
  __syncthreads();

  if (tid == 0) {
    float L_overall = s_tot[0] / s_tot[1];
    float L_peak    = s_tot[2] / ((float)DAYS * s_tot[3]);
    float L_time    = 10.0f * s_tot[4] / ((float)B * (float)DAYS);
    out[0] = 1.0f * L_overall + 2.0f * L_peak + 1.0f * L_time;
  }
}

extern "C" void kernel_launch(void* const* d_in, const int* in_sizes, int n_in,
                              void* d_out, int out_size, void* d_ws, size_t ws_size,
                              hipStream_t stream) {
  const float* y_pred = (const float*)d_in[0];
  const float* y_true = (const float*)d_in[1];
  const int B = in_sizes[0] / SEQ;   // 4096

  float* ws    = (float*)d_ws;       // 5 * B floats = 80 KB scratch
  float* o_wsq = ws;
  float* o_ws  = ws + (size_t)B;
  float* o_psq = ws + (size_t)2 * B;
  float* o_avg = ws + (size_t)3 * B;
  float* o_dt  = ws + (size_t)4 * B;

  wpal_per_sample<<<B, BLK1, 0, stream>>>(y_pred, y_true,
                                          o_wsq, o_ws, o_psq, o_avg, o_dt);
  wpal_finalize<<<1, BLK2, 0, stream>>>(o_wsq, o_ws, o_psq, o_avg, o_dt,
                                        (float*)d_out, B);
}